// ChamferDistance_39865886442283
// MI455X (gfx1250) — compile-verified
//
#include <hip/hip_runtime.h>
#include <hip/hip_bf16.h>

typedef __attribute__((ext_vector_type(2))) float v2f;
typedef __attribute__((ext_vector_type(8))) float v8f;

#define CH_B 16
#define CH_N 4096

// ---------------------------------------------------------------------------
// Pass 0: L2-normalize points, store (x, y, z, ||p||^2) as padded float4.
// ---------------------------------------------------------------------------
__global__ void chamfer_normalize_kernel(const float* __restrict__ in,
                                         float4* __restrict__ out,
                                         int npts) {
    int i = blockIdx.x * blockDim.x + threadIdx.x;
    if (i >= npts) return;
    float x = in[3 * i + 0];
    float y = in[3 * i + 1];
    float z = in[3 * i + 2];
    float n = __builtin_sqrtf(x * x + y * y + z * z);   // precise (one-time cost)
    float d = fmaxf(n, 1e-12f);
    float inv = 1.0f / d;
    float nx = x * inv, ny = y * inv, nz = z * inv;
    float sq = nx * nx + ny * ny + nz * nz;
    out[i] = make_float4(nx, ny, nz, sq);
}

// ---------------------------------------------------------------------------
// Pass 1/2: for each "row" point, min distance over all "col" points.
// One wave32 owns 16 rows of one batch; loops over col tiles of 16.
// Dot products per 16x16 tile via V_WMMA_F32_16X16X4_F32 (K=4, z-padded).
//
// Inner loop tracks min *squared* distance; sqrt(max(.,0)) is monotonic so it
// commutes with min and is applied once per row after the loop.
//
// A layout (16x4 f32, ISA 7.12.2): lanes 0-15 hold M=lane:    VGPR0=K0, VGPR1=K1
//                                  lanes 16-31 hold M=lane-16: VGPR0=K2, VGPR1=K3(=0)
// B layout mirrored (lane = N).  C/D: lane = N (mod 16); VGPR v = row v
// (lanes 0-15) or row v+8 (lanes 16-31).
// ---------------------------------------------------------------------------
__global__ void chamfer_min_kernel(const float4* __restrict__ RowPts,  // B*N float4
                                   const float4* __restrict__ ColPts,  // B*M float4
                                   float* __restrict__ minOut,         // B*N
                                   int N, int M) {
    const int lane  = threadIdx.x & 31;
    const int wave  = threadIdx.x >> 5;
    const int wavesPerBlock = blockDim.x >> 5;
    const int gwave = blockIdx.x * wavesPerBlock + wave;
    const int wavesPerBatch = N >> 4;                 // 256
    const int b       = gwave / wavesPerBatch;
    const int rowbase = (gwave % wavesPerBatch) << 4; // 16 rows per wave

    const float4* __restrict__ Arow = RowPts + (size_t)b * N;
    const float4* __restrict__ Bcol = ColPts + (size_t)b * M;

    const int half = lane >> 4;   // 0: K0/K1 half, 1: K2/pad half
    const int l15  = lane & 15;

    // A-matrix registers (constant across the whole col loop)
    float4 ap = Arow[rowbase + l15];
    v2f a;
    a.x = half ? ap.z : ap.x;
    a.y = half ? 0.0f : ap.y;

    // squared norms of this wave's 16 rows (per-lane: the 8 rows its C half covers)
    float pn[8];
#pragma unroll
    for (int v = 0; v < 8; ++v)
        pn[v] = Arow[rowbase + v + (half ? 8 : 0)].w;

    float rmin[8];  // running min of squared distance
#pragma unroll
    for (int v = 0; v < 8; ++v) rmin[v] = 3.402823e38f;

    for (int m0 = 0; m0 < M; m0 += 16) {
        float4 tp = Bcol[m0 + l15];
        v2f bb;
        bb.x = half ? tp.z : tp.x;
        bb.y = half ? 0.0f : tp.y;
        float tn = tp.w;           // col for this lane is m0 + l15 in both halves

        v8f c = {};
        c = __builtin_amdgcn_wmma_f32_16x16x4_f32(
                /*neg_a=*/false, a, /*neg_b=*/false, bb,
                /*c_mod=*/(short)0, c, /*reuse_a=*/false, /*reuse_b=*/false);

#pragma unroll
        for (int v = 0; v < 8; ++v) {
            float d2 = (pn[v] + tn) - 2.0f * c[v];
            rmin[v] = fminf(rmin[v], d2);    // no clamp/sqrt in the hot loop
        }
    }

    // min across the 16 lanes of each half (xor butterfly stays within halves),
    // then clamp + single hardware sqrt per row.
#pragma unroll
    for (int v = 0; v < 8; ++v) {
        float r = rmin[v];
        r = fminf(r, __shfl_xor(r, 1, 32));
        r = fminf(r, __shfl_xor(r, 2, 32));
        r = fminf(r, __shfl_xor(r, 4, 32));
        r = fminf(r, __shfl_xor(r, 8, 32));
        rmin[v] = __builtin_amdgcn_sqrtf(fmaxf(r, 0.0f));   // bare v_sqrt_f32
    }

    float* outRow = minOut + (size_t)b * N + rowbase;
    if (lane == 0) {
#pragma unroll
        for (int v = 0; v < 8; ++v) outRow[v] = rmin[v];
    } else if (lane == 16) {
#pragma unroll
        for (int v = 0; v < 8; ++v) outRow[8 + v] = rmin[v];
    }
}

// ---------------------------------------------------------------------------
// Pass 3: sum all row/col minima, scale by 1/B (reduction = 'mean').
// ---------------------------------------------------------------------------
__global__ void chamfer_reduce_kernel(const float* __restrict__ vals, int n,
                                      float* __restrict__ out, float scale) {
    __shared__ float smem[256];
    float s = 0.0f;
    for (int i = threadIdx.x; i < n; i += blockDim.x) s += vals[i];
    smem[threadIdx.x] = s;
    __syncthreads();
    for (int off = 128; off > 0; off >>= 1) {
        if ((int)threadIdx.x < off) smem[threadIdx.x] += smem[threadIdx.x + off];
        __syncthreads();
    }
    if (threadIdx.x == 0) out[0] = smem[0] * scale;
}

extern "C" void kernel_launch(void* const* d_in, const int* in_sizes, int n_in,
                              void* d_out, int out_size, void* d_ws, size_t ws_size,
                              hipStream_t stream) {
    (void)in_sizes; (void)n_in; (void)out_size; (void)ws_size;

    const float* pred   = (const float*)d_in[0];
    const float* target = (const float*)d_in[1];
    float* out = (float*)d_out;

    const int B = CH_B, N = CH_N, M = CH_N;
    const int npts = B * N;

    // workspace layout
    float4* P4   = (float4*)d_ws;            // npts float4  (1 MB)
    float4* T4   = P4 + npts;                // npts float4  (1 MB)
    float*  minP = (float*)(T4 + npts);      // npts floats
    float*  minT = minP + npts;              // npts floats (contiguous with minP)

    // Pass 0: normalize both clouds
    {
        dim3 blk(256), grd((npts + 255) / 256);
        chamfer_normalize_kernel<<<grd, blk, 0, stream>>>(pred,   P4, npts);
        chamfer_normalize_kernel<<<grd, blk, 0, stream>>>(target, T4, npts);
    }

    // Pass 1: min over targets for each pred row  -> minP
    // Pass 2: min over preds  for each target row -> minT
    {
        const int wavesTotal = B * (N >> 4);         // 4096 waves
        const int wavesPerBlock = 8;                 // 256 threads
        dim3 blk(32 * wavesPerBlock), grd(wavesTotal / wavesPerBlock);
        chamfer_min_kernel<<<grd, blk, 0, stream>>>(P4, T4, minP, N, M);
        chamfer_min_kernel<<<grd, blk, 0, stream>>>(T4, P4, minT, M, N);
    }

    // Pass 3: total sum / B
    chamfer_reduce_kernel<<<1, 256, 0, stream>>>(minP, 2 * npts, out, 1.0f / (float)B);
}